// WarpAdjoint_12163347382514
// MI455X (gfx1250) — compile-verified
//
#include <hip/hip_runtime.h>

// Problem shape (fixed by setup_inputs)
#define B_   8
#define T_   12
#define M_   384
#define N_   384

// Tiling
#define TM   32
#define TN   32
#define HALO 8
#define LM   (TM + 2 * HALO)   // 48
#define LN   (TN + 2 * HALO)   // 48
#define NTHREADS 256           // 8 waves (wave32); each thread owns 4 consecutive pixels

__global__ __launch_bounds__(NTHREADS)
void zero_out_kernel(float* __restrict__ out, int n) {
    int i = blockIdx.x * NTHREADS + threadIdx.x;
    if (i < n) out[i] = 0.0f;
}

__global__ __launch_bounds__(NTHREADS)
void splat_sum_kernel(const float* __restrict__ x_re,
                      const float* __restrict__ x_im,
                      const float* __restrict__ u,   // [..., 2] interleaved (dx, dy)
                      float* __restrict__ out)       // [B, M, N] complex64 -> float2
{
    __shared__ float2 acc[LM * LN];   // 18432 B: tile + halo accumulator

    const int tid = threadIdx.x;
    const int b   = blockIdx.z;
    const int m0  = blockIdx.y * TM;
    const int n0  = blockIdx.x * TN;

    // Zero the LDS accumulator (ds_store_b64)
    for (int i = tid; i < LM * LN; i += NTHREADS)
        acc[i] = make_float2(0.0f, 0.0f);
    __syncthreads();

    const size_t plane = (size_t)M_ * (size_t)N_;

    // Each thread owns 4 consecutive pixels of the 32x32 tile:
    //   row  = tid / 8, cols = (tid % 8)*4 .. +3   -> b128-friendly
    const int lm_t = tid >> 3;          // 0..31
    const int ln_t = (tid & 7) * 4;     // 0,4,...,28
    const int m    = m0 + lm_t;
    const int nb   = n0 + ln_t;

    for (int t = 0; t < T_; ++t) {
        const size_t slice = (size_t)(b * T_ + t) * plane;
        const size_t idx   = slice + (size_t)m * N_ + nb;

        // Prefetch next T-slice of this thread's pixels into GL2 (global_prefetch_b8)
        if (t + 1 < T_) {
            __builtin_prefetch(&x_re[idx + plane], 0, 0);
            __builtin_prefetch(&x_im[idx + plane], 0, 0);
            __builtin_prefetch(&u[2 * (idx + plane)], 0, 0);
        }

        // Wide loads: 1x b128 re, 1x b128 im, 2x b128 flow pairs
        const float4 vr4 = *(const float4*)(x_re + idx);
        const float4 vi4 = *(const float4*)(x_im + idx);
        const float4 d01 = *(const float4*)(u + 2 * idx);       // (dx0,dy0,dx1,dy1)
        const float4 d23 = *(const float4*)(u + 2 * idx + 4);   // (dx2,dy2,dx3,dy3)

        const float vr[4] = {vr4.x, vr4.y, vr4.z, vr4.w};
        const float vi[4] = {vi4.x, vi4.y, vi4.z, vi4.w};
        const float dx[4] = {d01.x, d01.z, d23.x, d23.z};
        const float dy[4] = {d01.y, d01.w, d23.y, d23.w};

        #pragma unroll
        for (int j = 0; j < 4; ++j) {
            const int n = nb + j;

            const float fx  = (float)n + dx[j];
            const float fy  = (float)m + dy[j];
            const float x0f = floorf(fx);
            const float y0f = floorf(fy);
            const int   xi0 = (int)x0f;
            const int   yi0 = (int)y0f;
            const float ax  = fx - x0f;
            const float ay  = fy - y0f;
            const float wx[2] = {1.0f - ax, ax};
            const float wy[2] = {1.0f - ay, ay};

            #pragma unroll
            for (int cy = 0; cy < 2; ++cy) {
                const int yy = yi0 + cy;
                if (yy < 0 || yy >= M_) continue;
                #pragma unroll
                for (int cx = 0; cx < 2; ++cx) {
                    const int xx = xi0 + cx;
                    if (xx < 0 || xx >= N_) continue;
                    const float w   = wx[cx] * wy[cy];
                    const int   llm = yy - m0 + HALO;
                    const int   lln = xx - n0 + HALO;
                    if ((unsigned)llm < (unsigned)LM && (unsigned)lln < (unsigned)LN) {
                        // fast path: LDS atomics (ds_add_f32)
                        float* p = (float*)&acc[llm * LN + lln];
                        atomicAdd(p,     vr[j] * w);
                        atomicAdd(p + 1, vi[j] * w);
                    } else {
                        // rare out-of-halo fallback: HW global f32 atomic add
                        float* q = out + 2 * ((size_t)b * plane + (size_t)yy * N_ + xx);
                        unsafeAtomicAdd(q,     vr[j] * w);
                        unsafeAtomicAdd(q + 1, vi[j] * w);
                    }
                }
            }
        }
    }
    __syncthreads();

    // Flush tile + halo to global with global_atomic_add_f32
    for (int i = tid; i < LM * LN; i += NTHREADS) {
        const int lm = i / LN;
        const int ln = i - lm * LN;
        const int mm = m0 + lm - HALO;
        const int nn = n0 + ln - HALO;
        if ((unsigned)mm < (unsigned)M_ && (unsigned)nn < (unsigned)N_) {
            const float2 v = acc[i];
            if (v.x != 0.0f || v.y != 0.0f) {
                float* q = out + 2 * ((size_t)b * plane + (size_t)mm * N_ + nn);
                unsafeAtomicAdd(q,     v.x);
                unsafeAtomicAdd(q + 1, v.y);
            }
        }
    }
}

extern "C" void kernel_launch(void* const* d_in, const int* in_sizes, int n_in,
                              void* d_out, int out_size, void* d_ws, size_t ws_size,
                              hipStream_t stream) {
    const float* x_re = (const float*)d_in[0];
    const float* x_im = (const float*)d_in[1];
    const float* u    = (const float*)d_in[2];
    float* out = (float*)d_out;

    // Output is [B, M, N] complex64 == 2*B*M*N floats; must zero before scatter-add.
    const int n_out_floats = B_ * M_ * N_ * 2;
    zero_out_kernel<<<(n_out_floats + NTHREADS - 1) / NTHREADS, NTHREADS, 0, stream>>>(
        out, n_out_floats);

    dim3 grid(N_ / TN, M_ / TM, B_);
    splat_sum_kernel<<<grid, NTHREADS, 0, stream>>>(x_re, x_im, u, out);
}